// CFConv_71408126263663
// MI455X (gfx1250) — compile-verified
//
#include <hip/hip_runtime.h>
#include <cstdint>

// ---------------------------------------------------------------------------
// CFConv: y[idx_i[e]] += x[idx_j[e]] * Wij[e]   (idx_i sorted, F = 64)
// Bandwidth-bound (AI ~= 0.5 FLOP/B; HBM floor ~16 us @ 23.3 TB/s).
//  - Wij (320 MB, zero reuse): CDNA5 async global->LDS double buffering,
//    non-temporal hint so the stream does not evict L2-resident x.
//  - x (25.6 MB): fits in 192 MB L2; gathers use default RT policy.
//  - idx_i sorted: accumulate in registers per segment, flush with
//    global_atomic_add_f32 only at segment boundaries (~27 MB atomics).
// ---------------------------------------------------------------------------

namespace {
constexpr int kFeat          = 64;
constexpr int kWavesPerBlock = 4;
constexpr int kBlock         = 32 * kWavesPerBlock;      // 128 threads
constexpr int kTileE         = 16;                       // edges per LDS tile
constexpr int kTileFloats    = kTileE * kFeat;           // 1024 floats = 4 KB
constexpr int kAsyncPerTile  = kTileFloats / (32 * 4);   // 8 x b128 per wave
constexpr int kTilesPerWave  = 16;                       // 256 edges per wave
}

__global__ void cfconv_zero(float* __restrict__ y, int n4) {
  int i = blockIdx.x * blockDim.x + threadIdx.x;
  if (i < n4) reinterpret_cast<float4*>(y)[i] = float4{0.f, 0.f, 0.f, 0.f};
}

__global__ __launch_bounds__(kBlock)
void cfconv_kernel(const float* __restrict__ x,
                   const float* __restrict__ Wij,
                   const int*  __restrict__ idx_i,
                   const int*  __restrict__ idx_j,
                   float*      __restrict__ y,
                   int nEdges, int nTiles)
{
  __shared__ float lds[kWavesPerBlock][2][kTileFloats];

  const int lane   = threadIdx.x & 31;
  const int warp   = threadIdx.x >> 5;
  const int waveId = blockIdx.x * kWavesPerBlock + warp;
  const int t0     = waveId * kTilesPerWave;
  if (t0 >= nTiles) return;
  const int tEnd   = min(t0 + kTilesPerWave, nTiles);

  const long maxF4 = (long)nEdges * kFeat - 4;  // last valid float4 in Wij

  // Issue one 4 KB tile of Wij into LDS buffer `buf` with async copies.
  // Non-temporal: Wij is a one-pass 320 MB stream; keep it out of L2 so the
  // x gather table stays resident.
  auto issue = [&](int tile, int buf) {
    const long     base = (long)tile * kTileFloats;
    const uint32_t ldsb = (uint32_t)(uintptr_t)(&lds[warp][buf][0]);
#pragma unroll
    for (int k = 0; k < kAsyncPerTile; ++k) {
      long fo = base + (long)(k * 32 + lane) * 4;
      if (fo > maxF4) fo = maxF4;               // clamp (values masked later)
      const float* g = Wij + fo;
      uint32_t     l = ldsb + (uint32_t)((k * 32 + lane) * 16);
      asm volatile("global_load_async_to_lds_b128 %0, %1, off th:TH_LOAD_NT"
                   :: "v"(l), "v"(g) : "memory");
    }
  };

  issue(t0, 0);

  int   cur = -1;
  float ax = 0.f, ay = 0.f;
  const int f0 = 2 * lane;                      // this lane's feature pair

  for (int t = t0; t < tEnd; ++t) {
    const int  buf     = (t - t0) & 1;
    const bool hasNext = (t + 1) < tEnd;
    if (hasNext) issue(t + 1, buf ^ 1);         // prefetch next tile

    // One index load per tile: lanes 0..15 -> idx_i, lanes 16..31 -> idx_j.
    long eL = (long)t * kTileE + (lane & 15);
    if (eL >= nEdges) eL = nEdges - 1;
    const int raw = (lane < 16) ? idx_i[eL] : idx_j[eL];

    // Wait for current tile (in-order completion: <=8 left means tile t done).
    if (hasNext) asm volatile("s_wait_asynccnt 8" ::: "memory");
    else         asm volatile("s_wait_asynccnt 0" ::: "memory");

    const float2* lb = reinterpret_cast<const float2*>(&lds[warp][buf][0]);
    const float2* x2 = reinterpret_cast<const float2*>(x);

#pragma unroll
    for (int u = 0; u < kTileE; ++u) {
      const long e = (long)t * kTileE + u;
      if (e < nEdges) {
        const int vi = __shfl(raw, u, 32);       // segment (center atom)
        const int vj = __shfl(raw, 16 + u, 32);  // neighbor atom
        if (vi != cur) {                         // segment boundary: flush
          if (cur >= 0) {
            atomicAdd(&y[(long)cur * kFeat + f0],     ax);
            atomicAdd(&y[(long)cur * kFeat + f0 + 1], ay);
          }
          ax = 0.f; ay = 0.f; cur = vi;
        }
        const float2 w  = lb[u * 32 + lane];            // ds_load_b64
        const float2 xv = x2[(long)vj * 32 + lane];     // global_load_b64 (L2)
        ax = fmaf(w.x, xv.x, ax);
        ay = fmaf(w.y, xv.y, ay);
      }
    }
  }

  if (cur >= 0) {
    atomicAdd(&y[(long)cur * kFeat + f0],     ax);
    atomicAdd(&y[(long)cur * kFeat + f0 + 1], ay);
  }
}

extern "C" void kernel_launch(void* const* d_in, const int* in_sizes, int n_in,
                              void* d_out, int out_size, void* d_ws, size_t ws_size,
                              hipStream_t stream)
{
  const float* x     = (const float*)d_in[0];
  const float* Wij   = (const float*)d_in[1];
  const int*   idx_i = (const int*)d_in[2];
  const int*   idx_j = (const int*)d_in[3];
  float*       y     = (float*)d_out;

  const int nEdges = in_sizes[2];

  // Zero the (poisoned) output first.
  const int n4 = out_size / 4;
  cfconv_zero<<<(n4 + 255) / 256, 256, 0, stream>>>(y, n4);

  const int nTiles  = (nEdges + kTileE - 1) / kTileE;
  const int nWaves  = (nTiles + kTilesPerWave - 1) / kTilesPerWave;
  const int nBlocks = (nWaves + kWavesPerBlock - 1) / kWavesPerBlock;
  cfconv_kernel<<<nBlocks, kBlock, 0, stream>>>(x, Wij, idx_i, idx_j, y,
                                                nEdges, nTiles);
}